// DeformationLoss_644245095002
// MI455X (gfx1250) — compile-verified
//
#include <hip/hip_runtime.h>

typedef __attribute__((ext_vector_type(2))) float v2f;
typedef __attribute__((ext_vector_type(8))) float v8f;

#define BATCH 512
#define NJ    32
#define JOINTS_PER_WAVE 4
#define JGROUPS (NJ / JOINTS_PER_WAVE)           // 8
#define TILES_PER_DIM 32                         // 512 / 16
#define NTILES (TILES_PER_DIM * TILES_PER_DIM)   // 1024
#define NTASKS (NTILES * JGROUPS)                // 8192 wave-tasks
#define WAVES_PER_BLOCK 8
#define NBLOCKS (NTASKS / WAVES_PER_BLOCK)       // 1024

__device__ __forceinline__ float swz_add(float v, const int imm_result) {
    return v + __builtin_bit_cast(float, imm_result);
}

__device__ __forceinline__ float wave32_xor_add(float v) {
    // Butterfly reduction with ds_swizzle group-of-32 XOR mode:
    // offset[15]=0, and_mask=offset[4:0]=0x1F, or_mask=0, xor_mask=offset[14:10].
    // Builtin needs literal immediates, so unroll manually.
    v += __builtin_bit_cast(float,
         __builtin_amdgcn_ds_swizzle(__builtin_bit_cast(int, v), 0x401F)); // xor 16
    v += __builtin_bit_cast(float,
         __builtin_amdgcn_ds_swizzle(__builtin_bit_cast(int, v), 0x201F)); // xor 8
    v += __builtin_bit_cast(float,
         __builtin_amdgcn_ds_swizzle(__builtin_bit_cast(int, v), 0x101F)); // xor 4
    v += __builtin_bit_cast(float,
         __builtin_amdgcn_ds_swizzle(__builtin_bit_cast(int, v), 0x081F)); // xor 2
    v += __builtin_bit_cast(float,
         __builtin_amdgcn_ds_swizzle(__builtin_bit_cast(int, v), 0x041F)); // xor 1
    return v;
}

// Each wave computes a 16x16 tile of the pairwise-distance matrix for 4 joints
// (fully unrolled for load ILP), one V_WMMA_F32_16X16X4_F32 per joint:
//   A row (i side)  = [ x_i,  y_i,  z_i, |d_i|^2 ]
//   B col (j side)  = [-2x_j, -2y_j, -2z_j, 1    ]
//   D[m][n]         = |d_i|^2 - 2<d_i, d_j>
// The remaining |d_j|^2 is the lane's OWN column norm (element (m, n=lane&15)
// lives in the lane that loaded column n) -> no cross-lane traffic.
__global__ __launch_bounds__(256) void deform_tile_kernel(
    const float* __restrict__ pred, const float* __restrict__ recon,
    float* __restrict__ partial)
{
    const int lane = threadIdx.x & 31;
    const int wave = threadIdx.x >> 5;
    const int task = blockIdx.x * WAVES_PER_BLOCK + wave;   // 0..8191
    const int tile = task >> 3;                             // 0..1023
    const int jg   = task & 7;                              // joint group
    const int ti = tile >> 5;            // tile row (i index / 16)
    const int tj = tile & 31;            // tile col (j index / 16)
    const int m  = lane & 15;
    const int hi = lane >> 4;            // 0: lanes 0-15 hold {K0,K1}; 1: lanes 16-31 hold {K2,K3}

    const int iRow = ti * 16 + m;        // A-side batch index held by this lane
    const int iCol = tj * 16 + m;        // B-side batch index held by this lane
    const int kBase = jg * JOINTS_PER_WAVE;

    // Base pointers for this lane's row/col point at joint kBase; consecutive
    // joints are +3 floats -> compiler folds into immediate offsets.
    const float* prB = pred  + ((size_t)iRow * NJ + kBase) * 3;
    const float* rrB = recon + ((size_t)iRow * NJ + kBase) * 3;
    const float* pcB = pred  + ((size_t)iCol * NJ + kBase) * 3;
    const float* rcB = recon + ((size_t)iCol * NJ + kBase) * 3;

    float acc = 0.0f;

    #pragma unroll
    for (int k = 0; k < JOINTS_PER_WAVE; ++k) {
        float rx = prB[k * 3 + 0] - rrB[k * 3 + 0];
        float ry = prB[k * 3 + 1] - rrB[k * 3 + 1];
        float rz = prB[k * 3 + 2] - rrB[k * 3 + 2];
        float rn = rx * rx + ry * ry + rz * rz;      // |d_i|^2 (row norm)

        float cx = pcB[k * 3 + 0] - rcB[k * 3 + 0];
        float cy = pcB[k * 3 + 1] - rcB[k * 3 + 1];
        float cz = pcB[k * 3 + 2] - rcB[k * 3 + 2];
        float cn = cx * cx + cy * cy + cz * cz;      // |d_j|^2 (column norm)

        // 16x4 f32 A fragment: lo lanes {K0,K1} = {x_i, y_i}; hi lanes {K2,K3} = {z_i, |d_i|^2}
        v2f a, b;
        a.x = hi ? rz : rx;
        a.y = hi ? rn : ry;
        // 4x16 f32 B fragment: lo lanes {K0,K1} = {-2x_j, -2y_j}; hi lanes {K2,K3} = {-2z_j, 1}
        b.x = hi ? (-2.0f * cz) : (-2.0f * cx);
        b.y = hi ? 1.0f : (-2.0f * cy);

        v8f d = {};
        d = __builtin_amdgcn_wmma_f32_16x16x4_f32(
                /*neg_a=*/false, a, /*neg_b=*/false, b,
                /*c_mod=*/(short)0, d, /*reuse_a=*/false, /*reuse_b=*/false);

        // d[r] holds element (M = r + 8*hi, N = lane&15): |d_i|^2 - 2<d_i,d_j>.
        #pragma unroll
        for (int r = 0; r < 8; ++r) {
            float s = d[r] + cn;
            s = s > 0.0f ? s : 0.0f;                 // clamp round-off
            acc += __builtin_amdgcn_sqrtf(s);        // raw v_sqrt_f32
        }
    }

    acc = wave32_xor_add(acc);
    if (lane == 0)
        partial[task] = acc;
}

// Deterministic final reduction: sum 8192 partials, scale, write scalar loss.
__global__ __launch_bounds__(256) void deform_reduce_kernel(
    const float* __restrict__ partial, float* __restrict__ out)
{
    __shared__ float sm[256];
    float s = 0.0f;
    for (int i = threadIdx.x; i < NTASKS; i += 256)
        s += partial[i];
    sm[threadIdx.x] = s;
    __syncthreads();
    for (int stride = 128; stride > 0; stride >>= 1) {
        if (threadIdx.x < stride)
            sm[threadIdx.x] += sm[threadIdx.x + stride];
        __syncthreads();
    }
    if (threadIdx.x == 0) {
        const float num_pairs = (float)(BATCH * (BATCH - 1) / 2);   // 130816
        const float scale = 1.0f / (2.0f * (float)NJ * num_pairs);  // full matrix = 2x triangle
        out[0] = sm[0] * scale;
    }
}

extern "C" void kernel_launch(void* const* d_in, const int* in_sizes, int n_in,
                              void* d_out, int out_size, void* d_ws, size_t ws_size,
                              hipStream_t stream) {
    const float* pred  = (const float*)d_in[0];
    const float* recon = (const float*)d_in[1];
    float* partial = (float*)d_ws;   // NTASKS floats = 32 KB scratch

    deform_tile_kernel<<<NBLOCKS, 256, 0, stream>>>(pred, recon, partial);
    deform_reduce_kernel<<<1, 256, 0, stream>>>(partial, (float*)d_out);
}